// CBAM_46858093199833
// MI455X (gfx1250) — compile-verified
//
#include <hip/hip_runtime.h>
#include <math.h>

// CBAM for f[8,8,256,56,56] f32.  BT=64 frames, C=256, HW=3136.
// Bandwidth-bound: ~822 MB total traffic => ~35us at 23.3 TB/s.
// MLP (256->16->256) runs on the matrix pipe via V_WMMA_F32_16X16X4_F32.

typedef __attribute__((ext_vector_type(2))) float v2f;
typedef __attribute__((ext_vector_type(8))) float v8f;

#define BT   64
#define NC   256
#define CR   16
#define HWSZ 3136
#define HW4  784   // HW / 4 (float4 units)

__device__ __forceinline__ float sigmoidf_(float x) {
    return 1.0f / (1.0f + __expf(-x));
}

// ---------------------------------------------------------------------------
// Kernel A: per (bt,c) spatial max & mean over 56x56.  One block per (bt,c).
// ---------------------------------------------------------------------------
__global__ void cbam_pool(const float* __restrict__ f,
                          float* __restrict__ mx, float* __restrict__ av) {
    const int btc = blockIdx.x;                       // 0..BT*NC-1
    const float4* row = (const float4*)f + (size_t)btc * HW4;
    float vmax = -3.402823466e38f;
    float vsum = 0.0f;
    for (int i = threadIdx.x; i < HW4; i += 256) {
        float4 v = row[i];
        vmax = fmaxf(vmax, fmaxf(fmaxf(v.x, v.y), fmaxf(v.z, v.w)));
        vsum += (v.x + v.y) + (v.z + v.w);
    }
    __shared__ float smax[256];
    __shared__ float ssum[256];
    smax[threadIdx.x] = vmax;
    ssum[threadIdx.x] = vsum;
    __syncthreads();
    for (int s = 128; s > 0; s >>= 1) {
        if (threadIdx.x < s) {
            smax[threadIdx.x] = fmaxf(smax[threadIdx.x], smax[threadIdx.x + s]);
            ssum[threadIdx.x] += ssum[threadIdx.x + s];
        }
        __syncthreads();
    }
    if (threadIdx.x == 0) {
        mx[btc] = smax[0];
        av[btc] = ssum[0] * (1.0f / (float)HWSZ);
    }
}

// ---------------------------------------------------------------------------
// Kernel B: channel-attention MLP on the matrix pipe.
//   P = [mx; av]  (128 x 256)
//   H = relu(P @ w1)                (128 x 16)
//   Z = (H[0:64] + H[64:128]) @ w2  (64 x 256)   [second layer is linear]
//   ca = sigmoid(Z)
// One block of 256 threads = 8 wave32.  GEMM1: wave w owns rows 16w..16w+15.
// GEMM2: waves 0..3 own the 64 output rows.
// ---------------------------------------------------------------------------
__global__ void cbam_mlp_wmma(const float* __restrict__ mx,
                              const float* __restrict__ av,
                              const float* __restrict__ w1,
                              const float* __restrict__ w2,
                              float* __restrict__ ca) {
    const int tid = threadIdx.x;
    const int w  = tid >> 5;         // wave id (wave32)
    const int l  = tid & 31;         // lane id
    const int m  = l & 15;           // row-in-tile (A) / col-in-tile (B,D)
    const int kh = (l >> 4) << 1;    // lanes 16-31 hold K+2,K+3

    __shared__ float Hs[128][CR];    // H = relu(P@w1)

    // ---- GEMM1: 16x16 tile per wave, K = 256 in steps of 4 -----------------
    const float* Pw = (w < 4) ? (mx + (size_t)(16 * w) * NC)
                              : (av + (size_t)(16 * (w - 4)) * NC);
    v8f acc = {};
    for (int k = 0; k < NC; k += 4) {
        // A fragment: A[m][k+kh], A[m][k+kh+1]   (16x4 f32 layout)
        const float* ap = Pw + (size_t)m * NC + (k + kh);
        v2f a; a.x = ap[0]; a.y = ap[1];
        // B fragment: w1[k+kh][m], w1[k+kh+1][m] (4x16 f32 layout)
        const float* bp = w1 + (size_t)(k + kh) * CR + m;
        v2f b; b.x = bp[0]; b.y = bp[CR];
        acc = __builtin_amdgcn_wmma_f32_16x16x4_f32(
                  false, a, false, b, (short)0, acc, false, false);
    }
    // relu + park H tile in LDS (D layout: VGPR r -> rows r / r+8)
#pragma unroll
    for (int r = 0; r < 8; ++r) {
        int row = 16 * w + ((l < 16) ? r : r + 8);
        Hs[row][m] = fmaxf(acc[r], 0.0f);
    }
    __syncthreads();

    // ---- GEMM2: Z tile (16 rows) x 16 N-tiles, K = 16 in steps of 4 --------
    if (w < 4) {
        const int rowA = 16 * w + m;
        for (int n = 0; n < 16; ++n) {
            v8f acc2 = {};
#pragma unroll
            for (int kk = 0; kk < 4; ++kk) {
                const int kb = 4 * kk + kh;
                v2f a;
                a.x = Hs[rowA][kb]     + Hs[rowA + 64][kb];
                a.y = Hs[rowA][kb + 1] + Hs[rowA + 64][kb + 1];
                v2f b;
                b.x = w2[(size_t)kb * NC + 16 * n + m];
                b.y = w2[(size_t)(kb + 1) * NC + 16 * n + m];
                acc2 = __builtin_amdgcn_wmma_f32_16x16x4_f32(
                           false, a, false, b, (short)0, acc2, false, false);
            }
#pragma unroll
            for (int r = 0; r < 8; ++r) {
                int row = 16 * w + ((l < 16) ? r : r + 8);
                ca[(size_t)row * NC + 16 * n + m] = sigmoidf_(acc2[r]);
            }
        }
    }
}

// ---------------------------------------------------------------------------
// Kernel C: xc = ca*x; mean/max over C per pixel -> s[bt][2][HW].
// Grid (BT, 4): each thread owns one float4 of pixels, loops over 256 channels.
// ---------------------------------------------------------------------------
__global__ void cbam_spatial_stats(const float* __restrict__ f,
                                   const float* __restrict__ ca,
                                   float* __restrict__ s) {
    const int bt = blockIdx.x;
    const int p4 = blockIdx.y * 256 + threadIdx.x;
    __shared__ float cas[NC];
    cas[threadIdx.x] = ca[bt * NC + threadIdx.x];
    __syncthreads();
    if (p4 >= HW4) return;

    const float4* base = (const float4*)f + (size_t)bt * NC * HW4;
    float sx = 0.f, sy = 0.f, sz = 0.f, sw = 0.f;
    float mxx = -3.402823466e38f, mxy = mxx, mxz = mxx, mxw = mxx;
    for (int c = 0; c < NC; ++c) {
        float4 v = base[(size_t)c * HW4 + p4];
        const float a = cas[c];
        const float vx = v.x * a, vy = v.y * a, vz = v.z * a, vw = v.w * a;
        sx += vx; sy += vy; sz += vz; sw += vw;
        mxx = fmaxf(mxx, vx); mxy = fmaxf(mxy, vy);
        mxz = fmaxf(mxz, vz); mxw = fmaxf(mxw, vw);
    }
    const float inv = 1.0f / (float)NC;
    float4 avg4; avg4.x = sx * inv; avg4.y = sy * inv; avg4.z = sz * inv; avg4.w = sw * inv;
    float4 max4; max4.x = mxx; max4.y = mxy; max4.z = mxz; max4.w = mxw;
    ((float4*)(s + (size_t)bt * 2 * HWSZ))[p4]        = avg4;
    ((float4*)(s + (size_t)bt * 2 * HWSZ + HWSZ))[p4] = max4;
}

// ---------------------------------------------------------------------------
// Kernel D: 7x7 conv (2 in-ch -> 1), pad 3, + sigmoid -> sa[bt][HW].  Tiny.
// ---------------------------------------------------------------------------
__global__ void cbam_conv_sig(const float* __restrict__ s,
                              const float* __restrict__ cw,
                              float* __restrict__ sa) {
    const int bt = blockIdx.x;
    const int p  = blockIdx.y * 256 + threadIdx.x;
    __shared__ float wsh[98];
    if (threadIdx.x < 98) wsh[threadIdx.x] = cw[threadIdx.x];
    __syncthreads();
    if (p >= HWSZ) return;

    const int y = p / 56;
    const int x = p - y * 56;
    const float* sb = s + (size_t)bt * 2 * HWSZ;
    float acc = 0.0f;
#pragma unroll
    for (int ic = 0; ic < 2; ++ic) {
        for (int ky = 0; ky < 7; ++ky) {
            const int iy = y + ky - 3;
            if (iy < 0 || iy >= 56) continue;
#pragma unroll
            for (int kx = 0; kx < 7; ++kx) {
                const int ix = x + kx - 3;
                if (ix < 0 || ix >= 56) continue;
                acc += sb[ic * HWSZ + iy * 56 + ix] * wsh[ic * 49 + ky * 7 + kx];
            }
        }
    }
    sa[(size_t)bt * HWSZ + p] = sigmoidf_(acc);
}

// ---------------------------------------------------------------------------
// Kernel E: out = sa * ca * x, fully streamed in float4.
// ---------------------------------------------------------------------------
__global__ void cbam_final(const float* __restrict__ f,
                           const float* __restrict__ ca,
                           const float* __restrict__ sa,
                           float* __restrict__ out) {
    const unsigned g   = blockIdx.x * 256u + threadIdx.x;   // float4 index
    const unsigned bt  = g / (NC * HW4);
    const unsigned rem = g - bt * (NC * HW4);
    const unsigned c   = rem / HW4;
    const unsigned p4  = rem - c * HW4;

    const float a = ca[bt * NC + c];
    const float4 sv = ((const float4*)sa)[bt * HW4 + p4];
    const float4 v  = ((const float4*)f)[g];
    float4 o;
    o.x = v.x * a * sv.x;
    o.y = v.y * a * sv.y;
    o.z = v.z * a * sv.z;
    o.w = v.w * a * sv.w;
    ((float4*)out)[g] = o;
}

// ---------------------------------------------------------------------------
extern "C" void kernel_launch(void* const* d_in, const int* in_sizes, int n_in,
                              void* d_out, int out_size, void* d_ws, size_t ws_size,
                              hipStream_t stream) {
    const float* f  = (const float*)d_in[0];   // [8,8,256,56,56]
    const float* w1 = (const float*)d_in[1];   // [256,16]
    const float* w2 = (const float*)d_in[2];   // [16,256]
    const float* cw = (const float*)d_in[3];   // [1,2,7,7]
    float* out = (float*)d_out;

    float* ws = (float*)d_ws;
    float* mx = ws;                 // BT*NC          = 16384
    float* av = ws + 16384;         // BT*NC          = 16384
    float* ca = ws + 32768;         // BT*NC          = 16384
    float* s  = ws + 49152;         // BT*2*HW        = 401408
    float* sa = ws + 450560;        // BT*HW          = 200704

    cbam_pool         <<<dim3(BT * NC), 256, 0, stream>>>(f, mx, av);
    cbam_mlp_wmma     <<<dim3(1),       256, 0, stream>>>(mx, av, w1, w2, ca);
    cbam_spatial_stats<<<dim3(BT, 4),   256, 0, stream>>>(f, ca, s);
    cbam_conv_sig     <<<dim3(BT, 13),  256, 0, stream>>>(s, cw, sa);
    cbam_final        <<<dim3((BT * NC * HW4) / 256), 256, 0, stream>>>(f, ca, sa, out);
}